// GCN_63565515981075
// MI455X (gfx1250) — compile-verified
//
#include <hip/hip_runtime.h>
#include <hip/hip_bf16.h>

// ---------------- problem constants (match reference) ----------------
#define NN 51200      // nodes
#define FF 400        // input features
#define HH 256        // hidden
#define CC 2          // classes
#define EE 819200     // edges
#define GG 128        // graphs
#define NEG_SLOPE 0.01f

typedef float v2f  __attribute__((ext_vector_type(2)));
typedef float v4f  __attribute__((ext_vector_type(4)));
typedef float v8f  __attribute__((ext_vector_type(8)));

// order-preserving monotonic encoding of IEEE-754 float into u32
__device__ __forceinline__ unsigned fenc(float f) {
    unsigned u = __float_as_uint(f);
    return (u & 0x80000000u) ? ~u : (u | 0x80000000u);
}
__device__ __forceinline__ float fdec(unsigned e) {
    unsigned u = (e & 0x80000000u) ? (e & 0x7FFFFFFFu) : ~e;
    return __uint_as_float(u);
}

// ---------------- init: deg=1 (self loop), poolenc=0 ----------------
__global__ void k_init(unsigned* __restrict__ deg, unsigned* __restrict__ poolenc) {
    int i = blockIdx.x * 256 + threadIdx.x;
    if (i < NN) deg[i] = 1u;
    if (i < GG * HH) poolenc[i] = 0u;   // decodes below any real value
}

// ---------------- in-degree count over target nodes ----------------
__global__ void k_count(const int* __restrict__ ecol, unsigned* __restrict__ deg) {
    int e = blockIdx.x * 256 + threadIdx.x;
    if (e < EE) atomicAdd(&deg[ecol[e]], 1u);
}

// ---------------- dis = rsqrt(deg) ----------------
__global__ void k_dis(const unsigned* __restrict__ deg, float* __restrict__ dis) {
    int i = blockIdx.x * 256 + threadIdx.x;
    if (i < NN) dis[i] = rsqrtf((float)deg[i]);
}

// ---------------- h = x @ W  via V_WMMA_F32_16X16X4_F32 ----------------
// block = 256 threads = 8 waves; block computes rows [blk*128, +128) x all 256 cols.
// Wave w: rows [blk*128 + w*16, +16), 16 column tiles of 16.
__global__ __launch_bounds__(256) void k_gemm(const float* __restrict__ x,
                                              const float* __restrict__ W,
                                              float* __restrict__ h) {
    __shared__ float sW[16 * 256];              // 16 K-rows x 256 cols = 16 KB
    const int tid  = threadIdx.x;
    const int wave = tid >> 5;
    const int lane = tid & 31;
    const int nlo  = lane & 15;                 // M index for A / N index for B
    const int hi   = lane >> 4;                 // selects K pair (A/B), row half (C)
    const int row0 = blockIdx.x * 128 + wave * 16;

    const float* xp = x + (size_t)(row0 + nlo) * FF + 2 * hi;

    v8f acc[16];
    v8f vzero = {};
#pragma unroll
    for (int t = 0; t < 16; ++t) acc[t] = vzero;

    for (int kc = 0; kc < 25; ++kc) {           // 25 chunks x 16 K = 400
        __syncthreads();
        const float* wsrc = W + (size_t)kc * 4096;
#pragma unroll
        for (int i = 0; i < 4; ++i) {           // 256 thr x 4 x float4 = 4096 dwords
            *(v4f*)&sW[tid * 4 + i * 1024] = *(const v4f*)&wsrc[tid * 4 + i * 1024];
        }
        __syncthreads();
#pragma unroll
        for (int ks = 0; ks < 4; ++ks) {        // 4 WMMA K-steps of 4
            // A tile 16x4 f32: lane(M + 16*(K>=2)) holds K pair {kk, kk+1}
            v2f a = *(const v2f*)(xp + kc * 16 + ks * 4);
            const int kl = ks * 4 + 2 * hi;     // K offset within LDS slab
#pragma unroll
            for (int t = 0; t < 16; ++t) {
                v2f b;
                b.x = sW[kl * 256       + t * 16 + nlo];
                b.y = sW[kl * 256 + 256 + t * 16 + nlo];
                acc[t] = __builtin_amdgcn_wmma_f32_16x16x4_f32(
                    /*neg_a=*/false, a, /*neg_b=*/false, b,
                    /*c_mod=*/(short)0, acc[t],
                    /*reuse_a=*/false, /*reuse_b=*/false);
            }
        }
    }
    // C/D layout: vgpr r -> row (r + 8*hi), lane nlo -> col
#pragma unroll
    for (int t = 0; t < 16; ++t) {
        float* hp = h + (size_t)(row0 + 8 * hi) * HH + t * 16 + nlo;
#pragma unroll
        for (int r = 0; r < 8; ++r) hp[(size_t)r * HH] = acc[t][r];
    }
}

// ---------------- agg = b_gcn + (1/deg)*h  (bias + self-loop folded) ----------------
__global__ void k_seed(const float* __restrict__ h, const float* __restrict__ dis,
                       const float* __restrict__ bgcn, float* __restrict__ agg) {
    int i = blockIdx.x * 256 + threadIdx.x;     // each thread: 4 elements
    int n = i >> 6;
    int f = (i & 63) * 4;
    float d  = dis[n];
    float d2 = d * d;                           // = 1/deg = self-loop norm
    v4f hv = *(const v4f*)&h[(size_t)n * HH + f];
    v4f bv = *(const v4f*)&bgcn[f];
    v4f o;
    o.x = bv.x + d2 * hv.x;
    o.y = bv.y + d2 * hv.y;
    o.z = bv.z + d2 * hv.z;
    o.w = bv.w + d2 * hv.w;
    *(v4f*)&agg[(size_t)n * HH + f] = o;
}

// ---------------- edge scatter: agg[col] += dis[row]*dis[col]*h[row] ----------------
// one wave32 per edge, 8 floats/lane, b128 gathers + f32 atomics into L2-resident agg
__global__ __launch_bounds__(256) void k_edges(const int* __restrict__ erow,
                                               const int* __restrict__ ecol,
                                               const float* __restrict__ h,
                                               const float* __restrict__ dis,
                                               float* __restrict__ agg) {
    int gt   = blockIdx.x * 256 + threadIdx.x;
    int e    = gt >> 5;
    int lane = gt & 31;
    if (e >= EE) return;
    int r = erow[e];
    int c = ecol[e];
    float nrm = dis[r] * dis[c];
    const float* hp = h   + (size_t)r * HH + lane * 8;
    float*       ap = agg + (size_t)c * HH + lane * 8;
    v4f v0 = *(const v4f*)(hp);
    v4f v1 = *(const v4f*)(hp + 4);
    atomicAdd(ap + 0, nrm * v0.x);
    atomicAdd(ap + 1, nrm * v0.y);
    atomicAdd(ap + 2, nrm * v0.z);
    atomicAdd(ap + 3, nrm * v0.w);
    atomicAdd(ap + 4, nrm * v1.x);
    atomicAdd(ap + 5, nrm * v1.y);
    atomicAdd(ap + 6, nrm * v1.z);
    atomicAdd(ap + 7, nrm * v1.w);
}

// ---------------- LeakyReLU + per-graph max pool (monotonic u32 atomics) ----------------
__global__ void k_act_pool(const float* __restrict__ agg, const int* __restrict__ batch,
                           unsigned* __restrict__ poolenc) {
    int i = blockIdx.x * 256 + threadIdx.x;     // each thread: 4 elements
    int n = i >> 6;
    int f = (i & 63) * 4;
    int g = batch[n];
    v4f v = *(const v4f*)&agg[(size_t)n * HH + f];
    unsigned* pp = poolenc + (size_t)g * HH + f;
    float a0 = v.x >= 0.f ? v.x : NEG_SLOPE * v.x;
    float a1 = v.y >= 0.f ? v.y : NEG_SLOPE * v.y;
    float a2 = v.z >= 0.f ? v.z : NEG_SLOPE * v.z;
    float a3 = v.w >= 0.f ? v.w : NEG_SLOPE * v.w;
    atomicMax(pp + 0, fenc(a0));
    atomicMax(pp + 1, fenc(a1));
    atomicMax(pp + 2, fenc(a2));
    atomicMax(pp + 3, fenc(a3));
}

// ---------------- decode pool -> d_out[256 + j] ----------------
__global__ void k_pool_out(const unsigned* __restrict__ poolenc, float* __restrict__ out) {
    int j = blockIdx.x * 256 + threadIdx.x;
    if (j < GG * HH) out[CC * GG + j] = fdec(poolenc[j]);
}

// ---------------- logits = pool @ W_lin + b_lin -> d_out[0 .. 255] ----------------
__global__ void k_logits(const unsigned* __restrict__ poolenc,
                         const float* __restrict__ Wlin, const float* __restrict__ blin,
                         float* __restrict__ out) {
    int t = threadIdx.x;                        // 256 = 128 graphs x 2 classes
    int g = t >> 1;
    int c = t & 1;
    float s = blin[c];
    for (int k = 0; k < HH; ++k)
        s += fdec(poolenc[(size_t)g * HH + k]) * Wlin[k * CC + c];
    out[g * CC + c] = s;
}

extern "C" void kernel_launch(void* const* d_in, const int* in_sizes, int n_in,
                              void* d_out, int out_size, void* d_ws, size_t ws_size,
                              hipStream_t stream) {
    const float* x     = (const float*)d_in[0];          // [N,F]
    const int*   eidx  = (const int*)d_in[1];            // [2,E] flat
    const int*   batch = (const int*)d_in[2];            // [N]
    const float* Wgcn  = (const float*)d_in[3];          // [F,H]
    const float* bgcn  = (const float*)d_in[4];          // [H]
    const float* Wlin  = (const float*)d_in[5];          // [H,C]
    const float* blin  = (const float*)d_in[6];          // [C]
    float*       out   = (float*)d_out;                  // [G*C logits | G*H pool]

    const int* erow = eidx;
    const int* ecol = eidx + EE;

    // workspace carve-up
    float*    h       = (float*)d_ws;                    // N*H
    float*    agg     = h + (size_t)NN * HH;             // N*H
    unsigned* deg     = (unsigned*)(agg + (size_t)NN * HH); // N
    float*    dis     = (float*)(deg + NN);              // N
    unsigned* poolenc = (unsigned*)(dis + NN);           // G*H

    k_init  <<<(NN + 255) / 256, 256, 0, stream>>>(deg, poolenc);
    k_count <<<(EE + 255) / 256, 256, 0, stream>>>(ecol, deg);
    k_dis   <<<(NN + 255) / 256, 256, 0, stream>>>(deg, dis);
    k_gemm  <<<NN / 128, 256, 0, stream>>>(x, Wgcn, h);
    k_seed  <<<(NN * HH / 4) / 256, 256, 0, stream>>>(h, dis, bgcn, agg);
    k_edges <<<(EE * 32) / 256, 256, 0, stream>>>(erow, ecol, h, dis, agg);
    k_act_pool<<<(NN * HH / 4) / 256, 256, 0, stream>>>(agg, batch, poolenc);
    k_pool_out<<<(GG * HH + 255) / 256, 256, 0, stream>>>(poolenc, out);
    k_logits<<<1, 256, 0, stream>>>(poolenc, Wlin, blin, out);
}